// MemoryConsolidation_49065706389518
// MI455X (gfx1250) — compile-verified
//
#include <hip/hip_runtime.h>
#include <cstdint>
#include <cmath>

typedef float v2f __attribute__((ext_vector_type(2)));
typedef float v8f __attribute__((ext_vector_type(8)));

#define B_Q      1024
#define N_PAT    100000
#define D_DIM    1024
#define TOPK     32
#define SLICES   64
#define NS       1563           // ceil(N_PAT / SLICES)
#define STEPS    25             // ceil(NS / 64): 4 n-tiles of 16 per step
#define SCSTR    68             // score row stride (16 lanes -> distinct banks)
#define NEG_INF  (-3.402823466e38f)

// ---------------------------------------------------------------------------
// Kernel 1: scores = q @ X^T via V_WMMA_F32_16X16X4_F32 with 4-way N register
// tiling (4 independent accumulator chains -> no WMMA RAW stalls, A reuse x4).
// Grid: (64 query tiles, 16 slice groups), block 128 = 4 waves; each wave owns
// one N-slice and a streaming per-row top-32 list.
// ---------------------------------------------------------------------------
__global__ __launch_bounds__(128)
void hopfield_scores_topk(const float* __restrict__ query,
                          const float* __restrict__ patterns,
                          float* __restrict__ pvals,
                          int* __restrict__ pidx)
{
    // A fragments pre-packed in exact WMMA lane order:
    //   afrag[kstep][flane*2 + j] = Q[row = flane&15][4*kstep + 2*(flane>>4) + j]
    // -> each wave's A load is a linear, conflict-free ds_load_b64.
    __shared__ float afrag[256 * 64];                 // 64 KB
    __shared__ float sc[4][16 * SCSTR];               // per-wave 16x64 score block
    __shared__ float topval[4][16 * TOPK];
    __shared__ int   topidx[4][16 * TOPK];

    const int tid   = threadIdx.x;
    const int wave  = tid >> 5;
    const int lane  = tid & 31;
    const int qt    = blockIdx.x;                 // query tile (16 rows)
    const int slice = blockIdx.y * 4 + wave;      // 0..63

    // ---- build packed A fragments (8192 b64 slots, 64 per thread) ----
    const float* qbase = query + (size_t)qt * 16 * D_DIM;
    for (int i = tid; i < 256 * 32; i += 128) {
        const int kstep = i >> 5;
        const int fl    = i & 31;                 // fragment lane
        const int row   = fl & 15;
        const int k     = (kstep << 2) + ((fl >> 4) << 1);
        const float2 q2 = *(const float2*)(qbase + row * D_DIM + k);
        *(float2*)&afrag[(kstep << 6) + (fl << 1)] = q2;
    }
    if (lane < 16) {
        for (int j = 0; j < TOPK; ++j) {
            topval[wave][lane * TOPK + j] = NEG_INF;
            topidx[wave][lane * TOPK + j] = 0;
        }
    }
    __syncthreads();

    const int nbeg  = slice * NS;
    const int nlim  = min(nbeg + NS, N_PAT);
    const int khalf = (lane >> 4) << 1;           // K sub-offset for this lane half
    const int ncl   = lane & 15;                  // B column within a 16-wide tile

    float rmin = NEG_INF;                         // running min of this row's top-32
    int   rpos = 0;

    for (int t = 0; t < STEPS; ++t) {
        const int n0 = nbeg + (t << 6);           // 64 pattern columns per step

        // per-lane B row pointers for the 4 n-tiles (clamped; masked later)
        const float* bptr0 = patterns + (size_t)min(n0      + ncl, N_PAT - 1) * D_DIM + khalf;
        const float* bptr1 = patterns + (size_t)min(n0 + 16 + ncl, N_PAT - 1) * D_DIM + khalf;
        const float* bptr2 = patterns + (size_t)min(n0 + 32 + ncl, N_PAT - 1) * D_DIM + khalf;
        const float* bptr3 = patterns + (size_t)min(n0 + 48 + ncl, N_PAT - 1) * D_DIM + khalf;

        v8f c0 = {0.f,0.f,0.f,0.f,0.f,0.f,0.f,0.f};
        v8f c1 = c0, c2 = c0, c3 = c0;

        #pragma unroll 4
        for (int ks = 0; ks < 256; ++ks) {
            const v2f a  = *(const v2f*)&afrag[(ks << 6) + (lane << 1)]; // linear ds_load_b64
            const int ko = ks << 2;
            const v2f b0 = *(const v2f*)(bptr0 + ko);
            const v2f b1 = *(const v2f*)(bptr1 + ko);
            const v2f b2 = *(const v2f*)(bptr2 + ko);
            const v2f b3 = *(const v2f*)(bptr3 + ko);
            c0 = __builtin_amdgcn_wmma_f32_16x16x4_f32(false, a, false, b0, (short)0, c0, false, false);
            c1 = __builtin_amdgcn_wmma_f32_16x16x4_f32(false, a, false, b1, (short)0, c1, false, false);
            c2 = __builtin_amdgcn_wmma_f32_16x16x4_f32(false, a, false, b2, (short)0, c2, false, false);
            c3 = __builtin_amdgcn_wmma_f32_16x16x4_f32(false, a, false, b3, (short)0, c3, false, false);
        }

        // spill the 16x64 score block: C elem (row = v + 8*(lane>>4), col = lane&15)
        {
            const int rhi = (lane >> 4) << 3;
            #pragma unroll
            for (int v = 0; v < 8; ++v) {
                float* srow = &sc[wave][(rhi + v) * SCSTR + ncl];
                srow[ 0] = c0[v];
                srow[16] = c1[v];
                srow[32] = c2[v];
                srow[48] = c3[v];
            }
        }
        __syncthreads();

        // streaming top-32 maintenance: lanes 0..15 each own one query row
        if (lane < 16) {
            for (int col = 0; col < 64; ++col) {
                const int n = n0 + col;
                const float s = (n < nlim) ? sc[wave][lane * SCSTR + col] : NEG_INF;
                if (s > rmin) {
                    topval[wave][lane * TOPK + rpos] = s;
                    topidx[wave][lane * TOPK + rpos] = n;
                    rmin = topval[wave][lane * TOPK];
                    rpos = 0;
                    for (int j = 1; j < TOPK; ++j) {
                        const float tv = topval[wave][lane * TOPK + j];
                        if (tv < rmin) { rmin = tv; rpos = j; }
                    }
                }
            }
        }
        __syncthreads();
    }

    // ---- write this slice's partial top-32 per row ----
    if (lane < 16) {
        const int grow = qt * 16 + lane;
        float* vdst = pvals + ((size_t)grow * SLICES + slice) * TOPK;
        int*   idst = pidx  + ((size_t)grow * SLICES + slice) * TOPK;
        for (int j = 0; j < TOPK; ++j) {
            vdst[j] = topval[wave][lane * TOPK + j];
            idst[j] = topidx[wave][lane * TOPK + j];
        }
    }
}

// ---------------------------------------------------------------------------
// Kernel 2: per query row, merge 64x32 partial candidates -> global top-32,
// softmax, then retrieved[row] = sum_k w_k * patterns[idx_k].
// Grid: 1024 blocks x 256 threads.
// ---------------------------------------------------------------------------
__global__ __launch_bounds__(256)
void hopfield_merge_retrieve(const float* __restrict__ patterns,
                             const float* __restrict__ pvals,
                             const int* __restrict__ pidx,
                             float* __restrict__ out)
{
    __shared__ float cand[SLICES * TOPK];
    __shared__ int   cidx[SLICES * TOPK];
    __shared__ float redv[256];
    __shared__ int   redp[256];
    __shared__ float selv[TOPK];
    __shared__ int   seli[TOPK];
    __shared__ float w[TOPK];

    const int tid = threadIdx.x;
    const int row = blockIdx.x;
    const int NC  = SLICES * TOPK;  // 2048

    for (int i = tid; i < NC; i += 256) {
        cand[i] = pvals[(size_t)row * NC + i];
        cidx[i] = pidx [(size_t)row * NC + i];
    }
    __syncthreads();

    // 32 rounds of parallel argmax over 2048 candidates
    for (int it = 0; it < TOPK; ++it) {
        float bv = NEG_INF; int bp = 0;
        for (int j = tid; j < NC; j += 256) {
            const float v = cand[j];
            if (v > bv) { bv = v; bp = j; }
        }
        redv[tid] = bv; redp[tid] = bp;
        __syncthreads();
        for (int s = 128; s > 0; s >>= 1) {
            if (tid < s && redv[tid + s] > redv[tid]) {
                redv[tid] = redv[tid + s];
                redp[tid] = redp[tid + s];
            }
            __syncthreads();
        }
        if (tid == 0) {
            const int p = redp[0];
            selv[it] = redv[0];
            seli[it] = cidx[p];
            cand[p]  = NEG_INF;
        }
        __syncthreads();
    }

    // softmax over the 32 selected scores (beta = 1.0); tiny, single thread
    if (tid == 0) {
        float m = selv[0];
        for (int j = 1; j < TOPK; ++j) m = fmaxf(m, selv[j]);
        float sum = 0.f;
        for (int j = 0; j < TOPK; ++j) { const float e = __expf(selv[j] - m); w[j] = e; sum += e; }
        const float inv = 1.f / sum;
        for (int j = 0; j < TOPK; ++j) w[j] *= inv;
    }
    __syncthreads();

    // retrieved[row][d] = sum_k w[k] * patterns[seli[k]][d]; coalesced float4
    const int d0 = tid << 2;                      // 256 threads * 4 = 1024 = D
    float4 acc = {0.f, 0.f, 0.f, 0.f};
    for (int k = 0; k < TOPK; ++k) {
        const float wk = w[k];
        const float4 p = *(const float4*)(patterns + (size_t)seli[k] * D_DIM + d0);
        acc.x += wk * p.x; acc.y += wk * p.y; acc.z += wk * p.z; acc.w += wk * p.w;
    }
    *(float4*)(out + (size_t)row * D_DIM + d0) = acc;
}

// ---------------------------------------------------------------------------
extern "C" void kernel_launch(void* const* d_in, const int* in_sizes, int n_in,
                              void* d_out, int out_size, void* d_ws, size_t ws_size,
                              hipStream_t stream)
{
    const float* query    = (const float*)d_in[0];   // [1024, 1024] f32
    const float* patterns = (const float*)d_in[1];   // [100000, 1024] f32
    // d_in[2] = top_k scalar (compile-time fixed at 32)
    float* out = (float*)d_out;                      // [1024, 1024] f32

    // workspace: partial top-k lists [1024 rows][64 slices][32] (vals then idx)
    float* pvals = (float*)d_ws;
    int*   pidx  = (int*)((char*)d_ws + (size_t)B_Q * SLICES * TOPK * sizeof(float));

    dim3 g1(B_Q / 16, SLICES / 4);
    hipLaunchKernelGGL(hopfield_scores_topk, g1, dim3(128), 0, stream,
                       query, patterns, pvals, pidx);

    hipLaunchKernelGGL(hopfield_merge_retrieve, dim3(B_Q), dim3(256), 0, stream,
                       patterns, pvals, pidx, out);
}